// AEloss_17789754540200
// MI455X (gfx1250) — compile-verified
//
#include <hip/hip_runtime.h>
#include <math.h>

// Problem constants (from reference)
#define Bn 64
#define Nn 65536
#define Tn 8
#define Mn 30
#define Kn 17
#define MK (Mn * Kn)   // 510 joints per batch

typedef __attribute__((ext_vector_type(2))) float v2f;
typedef __attribute__((ext_vector_type(8))) float v8f;

__launch_bounds__(256)
__global__ void ae_loss_kernel(const float* __restrict__ tags,
                               const int* __restrict__ kps,
                               float* __restrict__ out)
{
    const int b   = blockIdx.x;
    const int tid = threadIdx.x;

    __shared__ float s_g[MK][Tn];     // gathered tag vectors      (16320 B)
    __shared__ float s_vf[MK];        // validity as float
    __shared__ float s_mean[32][Tn];  // person means, zero-padded to 32 rows
    __shared__ float s_cnt[Mn];       // valid joints per person
    __shared__ float s_n2[32];        // ||mean_m||^2 (zero-padded)
    __shared__ float s_hasf[32];      // person-present mask as float (zero-padded)
    __shared__ float s_red[256];      // pull reduction scratch
    __shared__ float s_n;             // number of present persons

    const float* tb = tags + (size_t)b * Nn * Tn;
    const int*   kb = kps  + (size_t)b * MK * 2;

    // ---- Phase 1: gather tag vectors for this batch into LDS (32B each) ----
    for (int j = tid; j < MK; j += 256) {
        int idx = kb[2 * j];
        int val = kb[2 * j + 1];
        s_vf[j] = (val > 0) ? 1.0f : 0.0f;
        const float4* src = (const float4*)(tb + (size_t)idx * Tn);
        float4 lo = src[0];
        float4 hi = src[1];
        s_g[j][0] = lo.x; s_g[j][1] = lo.y; s_g[j][2] = lo.z; s_g[j][3] = lo.w;
        s_g[j][4] = hi.x; s_g[j][5] = hi.y; s_g[j][6] = hi.z; s_g[j][7] = hi.w;
    }
    __syncthreads();

    // ---- Phase 2: per-person masked mean (240 threads: (m,t) pairs) ----
    if (tid < Mn * Tn) {
        int m = tid / Tn, t = tid % Tn;
        float sum = 0.0f, cnt = 0.0f;
        #pragma unroll
        for (int k = 0; k < Kn; ++k) {
            float vf = s_vf[m * Kn + k];
            sum += s_g[m * Kn + k][t] * vf;
            cnt += vf;
        }
        float sc = fmaxf(cnt, 1.0f);
        s_mean[m][t] = sum / sc;
        if (t == 0) s_cnt[m] = cnt;
    } else if (tid >= 240 && tid < 256) {
        int r = tid - 240;                 // zero pad rows 30,31
        s_mean[30 + (r >> 3)][r & 7] = 0.0f;
    }
    __syncthreads();

    // ---- Phase 3: row norms, presence mask, person count ----
    if (tid < 32) {
        float s = 0.0f;
        #pragma unroll
        for (int t = 0; t < Tn; ++t) { float v = s_mean[tid][t]; s += v * v; }
        s_n2[tid]   = s;
        s_hasf[tid] = (tid < Mn && s_cnt[tid] > 0.0f) ? 1.0f : 0.0f;
    }
    if (tid == 0) {
        float n = 0.0f;
        for (int m = 0; m < Mn; ++m) n += (s_cnt[m] > 0.0f) ? 1.0f : 0.0f;
        s_n = n;
    }
    __syncthreads();

    // ---- Phase 4: pull term (per-joint, then block tree reduction) ----
    float pacc = 0.0f;
    for (int j = tid; j < MK; j += 256) {
        int m = j / Kn;
        float sq = 0.0f;
        #pragma unroll
        for (int t = 0; t < Tn; ++t) {
            float d = s_g[j][t] - s_mean[m][t];
            sq += d * d;
        }
        sq *= (1.0f / (float)Tn);
        pacc += sq * s_vf[j] / fmaxf(s_cnt[m], 1.0f);
    }
    s_red[tid] = pacc;
    __syncthreads();
    for (int off = 128; off > 0; off >>= 1) {
        if (tid < off) s_red[tid] += s_red[tid + off];
        __syncthreads();
    }

    // ---- Phase 5: push term via FP32 WMMA Gram matrix (wave 0, branchless) ----
    if (tid < 32) {
        const int lane = tid;
        const int half = lane >> 4;    // which half-wave
        const int l16  = lane & 15;
        float push = 0.0f;
        const float n = s_n;

        // Hoist per-row norm/mask values into registers (fully unrolled ->
        // stays in VGPRs; no ds loads inside the masked accumulation).
        float n2i[2][8], hii[2][8];
        #pragma unroll
        for (int I = 0; I < 2; ++I) {
            #pragma unroll
            for (int r = 0; r < 8; ++r) {
                int irow = I * 16 + r + half * 8;
                n2i[I][r] = s_n2[irow];
                hii[I][r] = s_hasf[irow];
            }
        }

        #pragma unroll
        for (int I = 0; I < 2; ++I) {
            #pragma unroll
            for (int J = 0; J < 2; ++J) {
                v8f c = {};
                #pragma unroll
                for (int kc = 0; kc < 2; ++kc) {
                    // 16x4 f32 A layout: lane -> M=lane%16; VGPR0/1 hold
                    // K = kc*4 + half*2, +1.
                    int kb0 = kc * 4 + half * 2;
                    v2f a, bm;
                    a.x  = s_mean[I * 16 + l16][kb0];
                    a.y  = s_mean[I * 16 + l16][kb0 + 1];
                    // B = Mean^T: B[k][n] = Mean[J*16+n][k] -> same load pattern
                    bm.x = s_mean[J * 16 + l16][kb0];
                    bm.y = s_mean[J * 16 + l16][kb0 + 1];
                    c = __builtin_amdgcn_wmma_f32_16x16x4_f32(
                            false, a, false, bm, (short)0, c, false, false);
                }
                // C/D layout: VGPR r -> M = r (lanes 0-15) or r+8 (lanes 16-31)
                const int   jcol = J * 16 + l16;
                const float n2j  = s_n2[jcol];
                const float hj   = s_hasf[jcol];
                #pragma unroll
                for (int r = 0; r < 8; ++r) {
                    const int irow = I * 16 + r + half * 8;
                    float mask = hii[I][r] * hj;
                    mask = (irow == jcol) ? 0.0f : mask;   // v_cndmask, no branch
                    float d2 = n2i[I][r] + n2j - 2.0f * c[r];
                    push += mask * __expf(-d2);
                }
            }
        }
        // wave32 reduction
        #pragma unroll
        for (int off = 16; off > 0; off >>= 1)
            push += __shfl_xor(push, off, 32);

        if (lane == 0) {
            float res = 0.0f;
            if (n > 1.0f) {
                float denom = fmaxf(n * (n - 1.0f), 1.0f);
                res = push / denom * 0.5f;
            }
            out[2 * b + 0] = res;                        // push
            out[2 * b + 1] = s_red[0] / fmaxf(n, 1.0f);  // pull
        }
    }
}

extern "C" void kernel_launch(void* const* d_in, const int* in_sizes, int n_in,
                              void* d_out, int out_size, void* d_ws, size_t ws_size,
                              hipStream_t stream) {
    (void)in_sizes; (void)n_in; (void)d_ws; (void)ws_size; (void)out_size;
    const float* tags = (const float*)d_in[0];   // [B, N, T] float32
    const int*   kps  = (const int*)d_in[1];     // [B, M, K, 2] int32
    float*       out  = (float*)d_out;           // [B, 2] float32
    ae_loss_kernel<<<Bn, 256, 0, stream>>>(tags, kps, out);
}